// HamModule_73177652789892
// MI455X (gfx1250) — compile-verified
//
#include <hip/hip_runtime.h>
#include <math.h>

#define L_SITES 18
#define DIM (1 << L_SITES)
#define MK 64
#define BS 256
#define NBLK (DIM / BS)
#define NBOND (L_SITES - 1)

typedef __attribute__((ext_vector_type(2))) float v2f;
typedef __attribute__((ext_vector_type(8))) float v8f;

// ---------------- helpers ----------------

__device__ __forceinline__ float block_reduce(float val, float* sdata) {
  int tid = threadIdx.x;
  __syncthreads();
  sdata[tid] = val;
  __syncthreads();
  for (int s = BS >> 1; s > 0; s >>= 1) {
    if (tid < s) sdata[tid] += sdata[tid + s];
    __syncthreads();
  }
  return sdata[0];
}

__device__ __forceinline__ unsigned int mix32(unsigned int x) {
  x ^= x >> 16; x *= 0x7feb352dU;
  x ^= x >> 15; x *= 0x846ca68bU;
  x ^= x >> 16;
  return x;
}

// ---------------- init: fields, v_prev=0, v0 = gaussian (unnormalized) ----------------

__global__ void init_kernel(const float* __restrict__ B0p, const float* __restrict__ Bextp,
                            const float* __restrict__ phi,
                            float* __restrict__ Bx, float* __restrict__ Bz,
                            float* __restrict__ vrow0, float* __restrict__ vrow1,
                            float* __restrict__ partials) {
  __shared__ float sdata[BS];
  if (blockIdx.x == 0 && threadIdx.x < L_SITES) {
    float b0 = B0p[0], be = Bextp[0], ph = phi[threadIdx.x];
    Bx[threadIdx.x] = b0 * sinf(ph);
    Bz[threadIdx.x] = b0 * cosf(ph) + be;
  }
  int x = blockIdx.x * BS + threadIdx.x;
  vrow0[x] = 0.0f;  // v_{-1} = 0
  unsigned int h1 = mix32((unsigned int)x * 2654435761u + 0x9e3779b9u);
  unsigned int h2 = mix32((unsigned int)x * 0x85ebca6bu + 0xc2b2ae35u);
  float u1 = ((float)((h1 >> 8) + 1u)) * (1.0f / 16777217.0f);  // (0,1]
  float u2 = ((float)(h2 >> 8)) * (1.0f / 16777216.0f);         // [0,1)
  float g = sqrtf(-2.0f * logf(u1)) * cosf(6.28318530717958647f * u2);
  vrow1[x] = g;
  float ps = block_reduce(g * g, sdata);
  if (threadIdx.x == 0) partials[blockIdx.x] = ps;
}

// ---------------- matvec: w = H v ; partials = per-block sum of v*w ----------------
// Site i <-> bit (L-1-i). Bond couples adjacent bit pairs (p, p-1), p = 1..L-1.
// out[x] = [0.25*(#equal - #differ) + sum_i Bz_i*(bit_i?-0.5:+0.5)] * v[x]
//          + sum_i 0.5*Bx_i * v[x ^ bit_i]
//          + sum_{pairs differing} 0.5 * v[x ^ (both bits)]

__global__ void matvec_kernel(const float* __restrict__ v, float* __restrict__ w,
                              const float* __restrict__ Bx, const float* __restrict__ Bz,
                              float* __restrict__ partials) {
  __shared__ float sdata[BS];
  __shared__ float sBx[L_SITES], sBz[L_SITES];
  if (threadIdx.x < L_SITES) {
    sBx[threadIdx.x] = Bx[threadIdx.x];
    sBz[threadIdx.x] = Bz[threadIdx.x];
  }
  __syncthreads();
  int x = blockIdx.x * BS + threadIdx.x;
  float vx = v[x];
  int ndiff = __popc((x ^ (x >> 1)) & ((1 << (L_SITES - 1)) - 1));
  float diag = 0.25f * (float)(NBOND - 2 * ndiff);  // J1 = 1
  float acc = 0.0f;
#pragma unroll
  for (int i = 0; i < L_SITES; ++i) {
    int b = L_SITES - 1 - i;
    int bit = (x >> b) & 1;
    diag += sBz[i] * (bit ? -0.5f : 0.5f);
    acc += 0.5f * sBx[i] * v[x ^ (1 << b)];
  }
#pragma unroll
  for (int p = 1; p < L_SITES; ++p) {
    int differ = ((x >> p) ^ (x >> (p - 1))) & 1;
    if (differ) acc += 0.5f * v[x ^ (3 << (p - 1))];
  }
  acc += diag * vx;
  w[x] = acc;
  float ps = block_reduce(vx * acc, sdata);
  if (threadIdx.x == 0) partials[blockIdx.x] = ps;
}

// ---------------- update: alpha = sum(partialsA); w -= alpha*v + beta_prev*v_prev ----------------

__global__ void update_kernel(float* __restrict__ w, const float* __restrict__ v,
                              const float* __restrict__ vp,
                              const float* __restrict__ partialsA, float* __restrict__ partialsB,
                              float* __restrict__ alphaArr, const float* __restrict__ betaArr,
                              int s) {
  __shared__ float sdata[BS];
  float asum = 0.0f;
  for (int i = threadIdx.x; i < NBLK; i += BS) asum += partialsA[i];
  float alpha = block_reduce(asum, sdata);
  float bp = (s == 0) ? 0.0f : betaArr[s - 1];
  int x = blockIdx.x * BS + threadIdx.x;
  float wv = w[x] - alpha * v[x] - bp * vp[x];
  w[x] = wv;
  float ps = block_reduce(wv * wv, sdata);
  if (threadIdx.x == 0) {
    partialsB[blockIdx.x] = ps;
    if (blockIdx.x == 0) alphaArr[s] = alpha;
  }
}

// ---------------- normalize: beta = sqrt(sum(partials)); dst = w / max(beta,1e-12) ----------------

__global__ void normalize_kernel(const float* __restrict__ w, float* __restrict__ dst,
                                 const float* __restrict__ partials, float* __restrict__ betaOut) {
  __shared__ float sdata[BS];
  float asum = 0.0f;
  for (int i = threadIdx.x; i < NBLK; i += BS) asum += partials[i];
  float ss = block_reduce(asum, sdata);
  float beta = sqrtf(ss);
  float inv = 1.0f / fmaxf(beta, 1e-12f);
  int x = blockIdx.x * BS + threadIdx.x;
  dst[x] = w[x] * inv;
  if (threadIdx.x == 0 && blockIdx.x == 0) betaOut[0] = beta;
}

// ---------------- 64x64 symmetric tridiagonal eigensolver (tql2/QL, double, 1 thread) ----------------
// Emits a PRE-PADDED eigenvector block Upad[64][16]: cols 0..3 = ascending eigenvectors,
// cols 4..15 = 0. This makes the WMMA B operand branch-free (padding is data, not control flow).

__global__ void eigh_kernel(const float* __restrict__ alphaArr, const float* __restrict__ betaArr,
                            float* __restrict__ Upad, float* __restrict__ eigvals_out,
                            double* __restrict__ dD, double* __restrict__ eD,
                            double* __restrict__ ZD) {
  if (threadIdx.x != 0 || blockIdx.x != 0) return;
  const int n = MK;
  for (int i = 0; i < n; ++i) {
    dD[i] = (double)alphaArr[i];
    eD[i] = (i < n - 1) ? (double)betaArr[i] : 0.0;  // T offdiag = beta[0..62]
  }
  for (int k = 0; k < n * n; ++k) ZD[k] = 0.0;
  for (int i = 0; i < n; ++i) ZD[i * n + i] = 1.0;

  for (int l = 0; l < n; ++l) {
    int iter = 0;
    int m;
    do {
      for (m = l; m < n - 1; ++m) {
        double dd = fabs(dD[m]) + fabs(dD[m + 1]);
        if (fabs(eD[m]) <= 1e-15 * dd) break;
      }
      if (m != l) {
        if (iter++ == 60) break;
        double g = (dD[l + 1] - dD[l]) / (2.0 * eD[l]);
        double r = sqrt(g * g + 1.0);
        g = dD[m] - dD[l] + eD[l] / (g + (g >= 0.0 ? fabs(r) : -fabs(r)));
        double si = 1.0, c = 1.0, p = 0.0;
        for (int i = m - 1; i >= l; --i) {
          double f = si * eD[i];
          double b = c * eD[i];
          r = sqrt(f * f + g * g);
          eD[i + 1] = r;
          if (r == 0.0) { dD[i + 1] -= p; eD[m] = 0.0; goto next_sweep; }
          si = f / r; c = g / r;
          g = dD[i + 1] - p;
          r = (dD[i] - g) * si + 2.0 * c * b;
          p = si * r;
          dD[i + 1] = g + p;
          g = c * r - b;
          for (int k = 0; k < n; ++k) {
            f = ZD[k * n + i + 1];
            ZD[k * n + i + 1] = si * ZD[k * n + i] + c * f;
            ZD[k * n + i]     = c * ZD[k * n + i] - si * f;
          }
        }
        dD[l] -= p; eD[l] = g; eD[m] = 0.0;
      }
      next_sweep:;
    } while (m != l);
  }
  // ascending sort with eigenvector column swaps
  for (int i = 0; i < n - 1; ++i) {
    int kmin = i;
    for (int j = i + 1; j < n; ++j)
      if (dD[j] < dD[kmin]) kmin = j;
    if (kmin != i) {
      double t = dD[i]; dD[i] = dD[kmin]; dD[kmin] = t;
      for (int k = 0; k < n; ++k) {
        double tz = ZD[k * n + i]; ZD[k * n + i] = ZD[k * n + kmin]; ZD[k * n + kmin] = tz;
      }
    }
  }
  for (int m = 0; m < n; ++m)
    for (int j = 0; j < 16; ++j)
      Upad[m * 16 + j] = (j < 4) ? (float)ZD[m * n + j] : 0.0f;
  for (int k = 0; k < 4; ++k) eigvals_out[k] = (float)dD[k];
}

// ---------------- projection: eigvecs(d,k) = sum_m V[m][d] * U[m][k], via V_WMMA_F32_16X16X4_F32 ----
// One wave owns a 16(d) x 16(n) f32 tile (n padded 4->16); K=64 accumulated in 16 wmma steps.
// A(16x4): lanes 0-15 M=lane K={k,k+1}; lanes 16-31 M=lane-16 K={k+2,k+3} (ISA 7.12.2).
// B(4x16): row-striped: lanes 0-15 N=lane rows {k,k+1}; lanes 16-31 N=lane-16 rows {k+2,k+3}.
// C/D(16x16): VGPR r: lanes 0-15 -> (M=r, N=lane); lanes 16-31 -> (M=r+8, N=lane-16).
// B comes from LDS-staged pre-padded Upad -> no exec-masked loads in the WMMA loop.

__global__ void proj_kernel(const float* __restrict__ Vbase, const float* __restrict__ Upad,
                            float* __restrict__ out) {
  __shared__ float sU[MK * 16];
  for (int i = threadIdx.x; i < MK * 16; i += BS) sU[i] = Upad[i];
  __syncthreads();

  int gtid = blockIdx.x * blockDim.x + threadIdx.x;
  int wave = gtid >> 5;
  int lane = threadIdx.x & 31;
  int d0 = wave * 16;
  int row = lane & 15;                 // M for A, N for B
  int koff = (lane >> 4) << 1;         // 0 for lanes 0-15, 2 for lanes 16-31

  // per-lane base pointer: V row m is stored at slot m+1
  const float* ap = Vbase + (size_t)(koff + 1) * DIM + d0 + row;
  const float* bp = sU + koff * 16 + row;

  v8f c = {0.f, 0.f, 0.f, 0.f, 0.f, 0.f, 0.f, 0.f};
#pragma unroll
  for (int kk = 0; kk < MK; kk += 4) {
    v2f a, b;
    a.x = ap[(size_t)kk * DIM];
    a.y = ap[(size_t)(kk + 1) * DIM];
    b.x = bp[kk * 16];
    b.y = bp[(kk + 1) * 16];
    c = __builtin_amdgcn_wmma_f32_16x16x4_f32(
        /*neg_a=*/false, a, /*neg_b=*/false, b,
        /*c_mod=*/(short)0, c, /*reuse_a=*/false, /*reuse_b=*/false);
  }
  union { v8f v; float f[8]; } cu; cu.v = c;
  int n = lane & 15;
  if (n < 4) {
    int mbase = (lane < 16) ? 0 : 8;
#pragma unroll
    for (int r = 0; r < 8; ++r)
      out[(size_t)(d0 + mbase + r) * 4 + n] = cu.f[r];
  }
}

// ---------------- host driver ----------------

extern "C" void kernel_launch(void* const* d_in, const int* in_sizes, int n_in,
                              void* d_out, int out_size, void* d_ws, size_t ws_size,
                              hipStream_t stream) {
  (void)in_sizes; (void)n_in; (void)out_size; (void)ws_size;
  const float* B0   = (const float*)d_in[0];
  const float* Bext = (const float*)d_in[1];
  const float* phi  = (const float*)d_in[2];
  // d_in[3] = n_eigs (always 4 per reference)

  char* ws = (char*)d_ws;
  size_t off = 0;
  auto alloc = [&](size_t bytes, size_t align) -> char* {
    off = (off + align - 1) & ~(align - 1);
    char* p = ws + off;
    off += bytes;
    return p;
  };
  // V layout: row 0 = v_{-1} (zeros), rows 1..65 = v_0..v_64
  float* V        = (float*)alloc((size_t)(MK + 2) * DIM * sizeof(float), 16);
  float* w        = (float*)alloc((size_t)DIM * sizeof(float), 16);
  float* partA    = (float*)alloc(NBLK * sizeof(float), 16);
  float* partB    = (float*)alloc(NBLK * sizeof(float), 16);
  float* Bx       = (float*)alloc(32 * sizeof(float), 16);
  float* Bz       = (float*)alloc(32 * sizeof(float), 16);
  float* alphaArr = (float*)alloc(MK * sizeof(float), 16);
  float* betaArr  = (float*)alloc(MK * sizeof(float), 16);
  float* Upad     = (float*)alloc((size_t)MK * 16 * sizeof(float), 16);
  float* dummy    = (float*)alloc(16 * sizeof(float), 16);
  double* dD      = (double*)alloc(MK * sizeof(double), 8);
  double* eD      = (double*)alloc(MK * sizeof(double), 8);
  double* ZD      = (double*)alloc((size_t)MK * MK * sizeof(double), 8);

  float* out = (float*)d_out;  // [0..3] eigvals, [4..] eigvecs (DIM x 4 row-major)

  init_kernel<<<NBLK, BS, 0, stream>>>(B0, Bext, phi, Bx, Bz, V, V + (size_t)DIM, partA);
  normalize_kernel<<<NBLK, BS, 0, stream>>>(V + (size_t)DIM, V + (size_t)DIM, partA, dummy);

  for (int s = 0; s < MK; ++s) {
    const float* v  = V + (size_t)(s + 1) * DIM;
    const float* vp = V + (size_t)s * DIM;
    float* vn       = V + (size_t)(s + 2) * DIM;
    matvec_kernel<<<NBLK, BS, 0, stream>>>(v, w, Bx, Bz, partA);
    update_kernel<<<NBLK, BS, 0, stream>>>(w, v, vp, partA, partB, alphaArr, betaArr, s);
    normalize_kernel<<<NBLK, BS, 0, stream>>>(w, vn, partB, betaArr + s);
  }

  eigh_kernel<<<1, 1, 0, stream>>>(alphaArr, betaArr, Upad, out, dD, eD, ZD);

  int proj_blocks = (DIM / 16) * 32 / BS;  // 16384 waves, 8 waves/block -> 2048 blocks
  proj_kernel<<<proj_blocks, BS, 0, stream>>>(V, Upad, out + 4);
}